// NeuralNetwork_18184891531448
// MI455X (gfx1250) — compile-verified
//
#include <hip/hip_runtime.h>
#include <cmath>
#include <vector>
#include <complex>
#include <cstring>
#include <algorithm>

#define FDIM  96
#define KRBF  32
#define NPATH 15
#define PF    (NPATH*FDIM)   // 1440
#define TE    16
#define NEDGE 160000
#define NNODE 10000
#define NLAYER 3
#define CH9   9              // 1 + 3 + 5 spherical channels

// ---------------- CG nonzero table (host-built, bucketed by output channel) --
struct CGTab {
  int ostart[10];     // segment per global output channel c in [0,9)
  int ea[1024];       // global a index (0..8) into xj
  int eb[1024];       // global b index (0..8) into Y
  int ep[1024];       // path index (0..14) into G
  float ev[1024];     // CG coefficient
};

typedef __attribute__((ext_vector_type(2))) float v2f;
typedef __attribute__((ext_vector_type(8))) float v8f;

__device__ __forceinline__ v8f wmma4(v2f a, v2f b, v8f c) {
  // D(16x16,f32) = A(16x4,f32) * B(4x16,f32) + C   — native fp32 WMMA on CDNA5
  return __builtin_amdgcn_wmma_f32_16x16x4_f32(false, a, false, b, (short)0, c,
                                               false, false);
}

// ---------------------------------------------------------------- kernels ---
__global__ __launch_bounds__(256) void geom_kernel(
    const float* __restrict__ pos, const int* __restrict__ idx_i,
    const int* __restrict__ idx_j, float* __restrict__ rbf,
    float* __restrict__ Yall, int E)
{
  int e = blockIdx.x * 256 + threadIdx.x;
  if (e >= E) return;
  int i = idx_i[e], j = idx_j[e];
  float vx = pos[j*3+0] - pos[i*3+0];
  float vy = pos[j*3+1] - pos[i*3+1];
  float vz = pos[j*3+2] - pos[i*3+2];
  float r = sqrtf(vx*vx + vy*vy + vz*vz);
  float inv = 1.0f / (r + 1e-9f);
  float dx = vx*inv, dy = vy*inv, dz = vz*inv;

  float* Y = Yall + (size_t)e * CH9;
  const float c0  = 0.28209479177387814f;   // 0.5/sqrt(pi)
  const float c1  = 0.48860251190291992f;   // sqrt(3/(4pi))
  const float c2a = 1.09254843059207907f;   // 0.5*sqrt(15/pi)
  const float c2b = 0.31539156525252005f;   // 0.25*sqrt(5/pi)
  const float c2c = 0.54627421529603959f;   // 0.25*sqrt(15/pi)
  Y[0] = c0;
  Y[1] = c1*dy; Y[2] = c1*dz; Y[3] = c1*dx;
  Y[4] = c2a*dx*dy; Y[5] = c2a*dy*dz; Y[6] = c2b*(3.0f*dz*dz - 1.0f);
  Y[7] = c2a*dx*dz; Y[8] = c2c*(dx*dx - dy*dy);

  const float PI = 3.14159265358979323846f;
  float fcut = (r < 5.0f) ? 0.5f * (cosf(PI * r / 5.0f) + 1.0f) : 0.0f;
  float er = expf(-r);
  const float e5 = 0.006737946999085467f;     // exp(-5)
  float alpha = 32.0f / (1.0f - e5); alpha *= alpha;
  float* R = rbf + (size_t)e * KRBF;
  #pragma unroll
  for (int k = 0; k < KRBF; ++k) {
    float mu = e5 + (1.0f - e5) * (float)k / 31.0f;
    float d = er - mu;
    R[k] = expf(-alpha * d * d) * fcut;
  }
}

__global__ __launch_bounds__(256) void xinit_kernel(
    const int* __restrict__ Z, const float* __restrict__ emb,
    float* __restrict__ x, int N)
{
  int i = blockIdx.x * 256 + threadIdx.x;
  if (i >= N * FDIM) return;
  int n = i / FDIM, f = i - n * FDIM;
  float* xp = x + (size_t)n * (CH9 * FDIM) + f;
  xp[0] = emb[(size_t)Z[n] * FDIM + f];
  #pragma unroll
  for (int a = 1; a < CH9; ++a) xp[a * FDIM] = 0.0f;
}

// generic small transpose: src (R x C) -> dst (C x R); makes WMMA B-fragment
// k-pairs contiguous so each step is one b64 load instead of two b32 loads.
__global__ __launch_bounds__(256) void transpose_kernel(
    const float* __restrict__ src, float* __restrict__ dst, int R, int C)
{
  int i = blockIdx.x * 256 + threadIdx.x;
  if (i >= R * C) return;
  int r = i / C, c = i - r * C;
  dst[(size_t)c * R + r] = src[i];
}

// Fused: G = rbf(16x32) * W_rad[m](32x1440) via fp32 WMMA into LDS,
// then CG tensor product + scatter-add with hardware float atomics.
__global__ __launch_bounds__(256) void msg_kernel(
    const float* __restrict__ x, const float* __restrict__ rbf,
    const float* __restrict__ Yall, const int* __restrict__ idx_i,
    const int* __restrict__ idx_j, const float* __restrict__ WradT_m, // [PF][KRBF]
    const CGTab* __restrict__ cg, float* __restrict__ agg)
{
  extern __shared__ char smraw[];
  float* G_s   = (float*)smraw;                 // TE*PF   (16*1440)
  float* rbf_s = G_s + TE * PF;                 // TE*KRBF
  float* Y_s   = rbf_s + TE * KRBF;             // TE*12 (padded 9)
  int*   ii_s  = (int*)(Y_s + TE * 12);         // TE
  int*   ij_s  = ii_s + TE;                     // TE

  const int t  = threadIdx.x;
  const int e0 = blockIdx.x * TE;

  // Prefetch the 16x9 gathered x-rows (3 cachelines each) so the random
  // gather in phase B overlaps with the WMMA GEMM in phase A.
  for (int i = t; i < TE * CH9 * 3; i += 256) {
    int el  = i / 27;
    int rem = i - el * 27;
    int a   = rem / 3;
    int seg = rem - a * 3;
    int nj  = idx_j[e0 + el];
    __builtin_prefetch(x + (size_t)nj * (CH9 * FDIM) + a * FDIM + seg * 32, 0, 1);
  }

  for (int i = t; i < TE * KRBF; i += 256)
    rbf_s[i] = rbf[(size_t)e0 * KRBF + i];            // [e_loc][k] row-major
  for (int i = t; i < TE * CH9; i += 256) {
    int e = i / CH9, a = i - e * CH9;
    Y_s[e * 12 + a] = Yall[(size_t)(e0 + e) * CH9 + a];
  }
  if (t < TE) { ii_s[t] = idx_i[e0 + t]; ij_s[t] = idx_j[e0 + t]; }
  __syncthreads();

  // ---- phase A: WMMA GEMM, 8 waves cover 90 column tiles -------------------
  const int w     = t >> 5;
  const int lane  = t & 31;
  const int mrow  = lane & 15;             // A row / D col
  const int khalf = (lane >> 4) << 1;      // k pair select (0 or 2)
  const int drow0 = (lane >> 4) << 3;      // D row base (0 or 8)
  for (int tile = w; tile < PF / 16; tile += 8) {
    const int col = tile * 16 + mrow;
    const float* bt = WradT_m + (size_t)col * KRBF;   // contiguous k per column
    v8f acc = {};
    #pragma unroll
    for (int step = 0; step < KRBF / 4; ++step) {
      const int k0 = step * 4 + khalf;                // even -> 8B aligned
      v2f av = *(const v2f*)(rbf_s + mrow * KRBF + k0);
      v2f bv = *(const v2f*)(bt + k0);
      acc = wmma4(av, bv, acc);
    }
    #pragma unroll
    for (int j = 0; j < 8; ++j)
      G_s[(drow0 + j) * PF + col] = acc[j];
  }
  __syncthreads();

  // ---- phase B: tensor product + scatter (x is L2-resident: 34.6MB < 192MB)
  for (int i = t; i < TE * FDIM; i += 256) {
    const int el = i / FDIM, f = i - el * FDIM;
    const int nj = ij_s[el], ni = ii_s[el];
    const float* xj = x + (size_t)nj * (CH9 * FDIM) + f;
    const float* Gl = G_s + el * PF + f;
    const float* Yl = Y_s + el * 12;
    float* ab = agg + (size_t)ni * (CH9 * FDIM) + f;
    #pragma unroll
    for (int c = 0; c < CH9; ++c) {
      float m = 0.0f;
      const int q1 = cg->ostart[c + 1];
      for (int q = cg->ostart[c]; q < q1; ++q)
        m += xj[cg->ea[q] * FDIM] * Yl[cg->eb[q]] * cg->ev[q]
             * Gl[cg->ep[q] * FDIM];
      atomicAdd(ab + c * FDIM, m);
    }
  }
}

// x[n,a,:] += agg[n,a,:] * W_self[m,l]   (rows = N*(2l+1), clean F x F GEMM)
__global__ __launch_bounds__(192) void self_kernel(
    const float* __restrict__ agg, const float* __restrict__ WlT, // [F][F] (g,f)
    float* __restrict__ x, int l)
{
  __shared__ float A_s[16][FDIM];
  const int base = blockIdx.x * 16;
  const int t = threadIdx.x;
  const int twol1 = 2 * l + 1, aoff = l * l;
  for (int i = t; i < 16 * FDIM; i += 192) {
    int r = i / FDIM, k = i - r * FDIM;
    int grow = base + r;
    int n = grow / twol1, a = aoff + (grow - n * twol1);
    A_s[r][k] = agg[(size_t)n * (CH9 * FDIM) + a * FDIM + k];
  }
  __syncthreads();
  const int w = t >> 5, lane = t & 31;
  const int mrow = lane & 15;
  const int col  = w * 16 + mrow;
  const int khalf = (lane >> 4) << 1;
  const float* bt = WlT + (size_t)col * FDIM;
  v8f acc = {};
  #pragma unroll
  for (int step = 0; step < FDIM / 4; ++step) {
    const int k0 = step * 4 + khalf;
    v2f av = *(const v2f*)(&A_s[mrow][k0]);
    v2f bv = *(const v2f*)(bt + k0);
    acc = wmma4(av, bv, acc);
  }
  const int drow0 = (lane >> 4) << 3;
  #pragma unroll
  for (int j = 0; j < 8; ++j) {
    int grow = base + drow0 + j;
    int n = grow / twol1, a = aoff + (grow - n * twol1);
    x[(size_t)n * (CH9 * FDIM) + a * FDIM + col] += acc[j];
  }
}

__global__ __launch_bounds__(256) void gate_kernel(float* __restrict__ x, int N)
{
  int i = blockIdx.x * 256 + threadIdx.x;
  if (i >= N * FDIM) return;
  int n = i / FDIM, f = i - n * FDIM;
  float* xp = x + (size_t)n * (CH9 * FDIM) + f;
  float g = 1.0f / (1.0f + expf(-xp[0]));
  #pragma unroll
  for (int a = 1; a < CH9; ++a) xp[a * FDIM] *= g;
}

// out[e, coff+a, :] = ((x_i + x_j) * Gp) * W_out[l]; Gp fused into A-tile.
__global__ __launch_bounds__(192) void out_kernel(
    const float* __restrict__ x, const float* __restrict__ rbf,
    const int* __restrict__ idx_i, const int* __restrict__ idx_j,
    const float* __restrict__ Wpair_l, const float* __restrict__ WoutT_l,
    float* __restrict__ out, int l)
{
  __shared__ float H_s[16][FDIM];
  const int base = blockIdx.x * 16;
  const int t = threadIdx.x;
  const int twol1 = 2 * l + 1, aoff = l * l;   // channel offset == l*l
  for (int i = t; i < 16 * FDIM; i += 192) {
    int r = i / FDIM, f = i - r * FDIM;
    int grow = base + r;
    int e = grow / twol1, a = aoff + (grow - e * twol1);
    float gp = 0.0f;
    const float* rb = rbf + (size_t)e * KRBF;
    #pragma unroll
    for (int k = 0; k < KRBF; ++k) gp += rb[k] * Wpair_l[k * FDIM + f];
    const int ni = idx_i[e], nj = idx_j[e];
    H_s[r][f] = (x[(size_t)ni * (CH9 * FDIM) + a * FDIM + f] +
                 x[(size_t)nj * (CH9 * FDIM) + a * FDIM + f]) * gp;
  }
  __syncthreads();
  const int w = t >> 5, lane = t & 31;
  const int mrow = lane & 15;
  const int col  = w * 16 + mrow;
  const int khalf = (lane >> 4) << 1;
  const float* bt = WoutT_l + (size_t)col * FDIM;
  v8f acc = {};
  #pragma unroll
  for (int step = 0; step < FDIM / 4; ++step) {
    const int k0 = step * 4 + khalf;
    v2f av = *(const v2f*)(&H_s[mrow][k0]);
    v2f bv = *(const v2f*)(bt + k0);
    acc = wmma4(av, bv, acc);
  }
  const int drow0 = (lane >> 4) << 3;
  #pragma unroll
  for (int j = 0; j < 8; ++j) {
    int grow = base + drow0 + j;
    int e = grow / twol1, c = aoff + (grow - e * twol1);
    out[(size_t)e * (CH9 * FDIM) + c * FDIM + col] = acc[j];
  }
}

// ---------------------- host-side Clebsch-Gordan construction ---------------
static double factd(int n) { double r = 1.0; for (int i = 2; i <= n; ++i) r *= i; return r; }

static double cg_complex(int j1, int m1, int j2, int m2, int j3, int m3) {
  if (m1 + m2 != m3) return 0.0;
  if (j1 + j2 - j3 < 0 || j1 - j2 + j3 < 0 || -j1 + j2 + j3 < 0) return 0.0;
  double pre = (2*j3+1) * factd(j3+j1-j2) * factd(j3-j1+j2) * factd(j1+j2-j3)
               / factd(j1+j2+j3+1);
  pre *= factd(j3+m3) * factd(j3-m3) * factd(j1-m1) * factd(j1+m1)
         * factd(j2-m2) * factd(j2+m2);
  double s = 0.0;
  for (int k = 0; k <= j1 + j2 - j3; ++k) {
    int d0=k, d1=j1+j2-j3-k, d2=j1-m1-k, d3=j2+m2-k, d4=j3-j2+m1+k, d5=j3-j1-m2+k;
    if (d0<0||d1<0||d2<0||d3<0||d4<0||d5<0) continue;
    double den = factd(d0)*factd(d1)*factd(d2)*factd(d3)*factd(d4)*factd(d5);
    s += ((k & 1) ? -1.0 : 1.0) / den;
  }
  return std::sqrt(pre) * s;
}

static void u_real(int l, std::complex<double> U[5][5]) {
  for (int a = 0; a < 5; ++a) for (int b = 0; b < 5; ++b) U[a][b] = 0.0;
  const double SQ2 = 1.0 / std::sqrt(2.0);
  U[l][l] = 1.0;
  for (int m = 1; m <= l; ++m) {
    double s = (m & 1) ? -SQ2 : SQ2;           // (-1)^m / sqrt(2)
    U[l+m][l+m] = s;
    U[l+m][l-m] = SQ2;
    U[l-m][l-m] = std::complex<double>(0.0, SQ2);
    U[l-m][l+m] = std::complex<double>(0.0, -s);
  }
}

static void build_cg(CGTab* T) {
  struct Ent { int c, a, b, p; double v; };
  std::vector<Ent> ents;
  const int aoffs[3] = {0, 1, 4};
  int p = 0;
  for (int l1 = 0; l1 <= 2; ++l1)
  for (int l2 = 0; l2 <= 2; ++l2)
  for (int l3 = std::abs(l1 - l2); l3 <= std::min(l1 + l2, 2); ++l3) {
    int n1 = 2*l1+1, n2 = 2*l2+1, n3 = 2*l3+1;
    double Cc[5][5][5] = {};
    for (int m1 = -l1; m1 <= l1; ++m1)
      for (int m2 = -l2; m2 <= l2; ++m2)
        for (int m3 = -l3; m3 <= l3; ++m3)
          Cc[m1+l1][m2+l2][m3+l3] = cg_complex(l1, m1, l2, m2, l3, m3);
    std::complex<double> U1[5][5], U2[5][5], U3[5][5];
    u_real(l1, U1); u_real(l2, U2); u_real(l3, U3);
    std::complex<double> Cr[5][5][5];
    double sre = 0.0, sim = 0.0;
    for (int a = 0; a < n1; ++a) for (int b = 0; b < n2; ++b)
      for (int c = 0; c < n3; ++c) {
        std::complex<double> s = 0.0;
        for (int m = 0; m < n1; ++m) for (int n = 0; n < n2; ++n)
          for (int o = 0; o < n3; ++o)
            s += std::conj(U1[a][m]) * std::conj(U2[b][n]) * U3[c][o]
                 * Cc[m][n][o];
        Cr[a][b][c] = s;
        sre += std::abs(s.real()); sim += std::abs(s.imag());
      }
    bool useim = sim > sre;
    double mx = 0.0;
    double Crr[5][5][5];
    for (int a = 0; a < n1; ++a) for (int b = 0; b < n2; ++b)
      for (int c = 0; c < n3; ++c) {
        Crr[a][b][c] = useim ? Cr[a][b][c].imag() : Cr[a][b][c].real();
        mx = std::max(mx, std::abs(Crr[a][b][c]));
      }
    if (mx <= 1e-8) continue;
    for (int a = 0; a < n1; ++a) for (int b = 0; b < n2; ++b)
      for (int c = 0; c < n3; ++c)
        if (std::abs(Crr[a][b][c]) > 1e-8 && ents.size() < 1024)
          ents.push_back({aoffs[l3]+c, aoffs[l1]+a, aoffs[l2]+b, p, Crr[a][b][c]});
    ++p;
  }
  std::memset(T, 0, sizeof(CGTab));
  int idx = 0;
  for (int c = 0; c < 9; ++c) {
    T->ostart[c] = idx;
    for (const Ent& e : ents)
      if (e.c == c && idx < 1024) {
        T->ea[idx] = e.a; T->eb[idx] = e.b; T->ep[idx] = e.p;
        T->ev[idx] = (float)e.v; ++idx;
      }
  }
  T->ostart[9] = idx;
}

static CGTab g_tab;   // static storage: valid at graph replay; rebuilt identically each call

// ------------------------------------------------------------------ launch --
extern "C" void kernel_launch(void* const* d_in, const int* in_sizes, int n_in,
                              void* d_out, int out_size, void* d_ws, size_t ws_size,
                              hipStream_t stream) {
  (void)in_sizes; (void)n_in; (void)out_size; (void)ws_size;
  const int*   Z     = (const int*)  d_in[0];
  const float* pos   = (const float*)d_in[1];
  const int*   idx_i = (const int*)  d_in[2];
  const int*   idx_j = (const int*)  d_in[3];
  const float* emb   = (const float*)d_in[4];
  const float* Wrad  = (const float*)d_in[5];
  const float* Wself = (const float*)d_in[6];
  const float* Wpair = (const float*)d_in[7];
  const float* Wout  = (const float*)d_in[8];
  float* out = (float*)d_out;

  char* ws = (char*)d_ws;
  size_t off = 0;
  CGTab* d_cg   = (CGTab*)(ws);          off += ((sizeof(CGTab) + 255) / 256) * 256;
  float* rbf    = (float*)(ws + off);    off += (size_t)NEDGE * KRBF * sizeof(float);
  float* Yb     = (float*)(ws + off);    off += (size_t)NEDGE * CH9 * sizeof(float);
  float* x      = (float*)(ws + off);    off += (size_t)NNODE * CH9 * FDIM * sizeof(float);
  float* agg    = (float*)(ws + off);    off += (size_t)NNODE * CH9 * FDIM * sizeof(float);
  float* WradT  = (float*)(ws + off);    off += (size_t)NLAYER * PF * KRBF * sizeof(float);
  float* WselfT = (float*)(ws + off);    off += (size_t)NLAYER * 3 * FDIM * FDIM * sizeof(float);
  float* WoutT  = (float*)(ws + off);    off += (size_t)3 * FDIM * FDIM * sizeof(float);

  build_cg(&g_tab);
  hipMemcpyAsync(d_cg, &g_tab, sizeof(CGTab), hipMemcpyHostToDevice, stream);

  // weight transposes (tiny; keeps WMMA B-fragment loads contiguous)
  for (int m = 0; m < NLAYER; ++m)
    transpose_kernel<<<(KRBF * PF + 255) / 256, 256, 0, stream>>>(
        Wrad + (size_t)m * KRBF * PF, WradT + (size_t)m * PF * KRBF, KRBF, PF);
  for (int i = 0; i < NLAYER * 3; ++i)
    transpose_kernel<<<(FDIM * FDIM + 255) / 256, 256, 0, stream>>>(
        Wself + (size_t)i * FDIM * FDIM, WselfT + (size_t)i * FDIM * FDIM, FDIM, FDIM);
  for (int l = 0; l < 3; ++l)
    transpose_kernel<<<(FDIM * FDIM + 255) / 256, 256, 0, stream>>>(
        Wout + (size_t)l * FDIM * FDIM, WoutT + (size_t)l * FDIM * FDIM, FDIM, FDIM);

  geom_kernel<<<(NEDGE + 255) / 256, 256, 0, stream>>>(pos, idx_i, idx_j, rbf, Yb, NEDGE);
  xinit_kernel<<<(NNODE * FDIM + 255) / 256, 256, 0, stream>>>(Z, emb, x, NNODE);

  const size_t msg_smem = ((size_t)TE * PF + TE * KRBF + TE * 12) * sizeof(float)
                          + (size_t)TE * 2 * sizeof(int);   // ~95 KB, fits 320KB WGP LDS
  for (int m = 0; m < NLAYER; ++m) {
    hipMemsetAsync(agg, 0, (size_t)NNODE * CH9 * FDIM * sizeof(float), stream);
    msg_kernel<<<NEDGE / TE, 256, msg_smem, stream>>>(
        x, rbf, Yb, idx_i, idx_j, WradT + (size_t)m * PF * KRBF, d_cg, agg);
    for (int l = 0; l < 3; ++l) {
      int rows = NNODE * (2 * l + 1);
      self_kernel<<<rows / 16, 192, 0, stream>>>(
          agg, WselfT + ((size_t)m * 3 + l) * FDIM * FDIM, x, l);
    }
    gate_kernel<<<(NNODE * FDIM + 255) / 256, 256, 0, stream>>>(x, NNODE);
  }
  for (int l = 0; l < 3; ++l) {
    int rows = NEDGE * (2 * l + 1);
    out_kernel<<<rows / 16, 192, 0, stream>>>(
        x, rbf, idx_i, idx_j, Wpair + (size_t)l * KRBF * FDIM,
        WoutT + (size_t)l * FDIM * FDIM, out, l);
  }
}